// KernelSVMRBF_59622736003634
// MI455X (gfx1250) — compile-verified
//
#include <hip/hip_runtime.h>

// ---------------------------------------------------------------------------
// SVM RBF prediction:  out[i] = sum_j exp(-g*||x_i - xt_j||^2) * (a_j*y_j) + b
// fp32-WMMA GEMM (V_WMMA_F32_16X16X4_F32) + fused RBF epilogue.
// Double-buffered ASYNC global->LDS staging (ASYNCcnt / s_wait_asynccnt).
// ---------------------------------------------------------------------------

#define GAMMA 0.5f
#define D_DIM 256
#define LDS_STRIDE 260      // 256 + 4 pad -> bank = (4*row + k) % 64, conflict free
#define ROWS_PER_BLOCK 32   // output rows per workgroup
#define CGRPS 4             // column groups (each shared by 2 waves)
#define TILE_N 16
#define TILE_FLOATS (CGRPS * TILE_N * LDS_STRIDE)   // one 4-tile buffer

typedef __attribute__((ext_vector_type(2))) float v2f;
typedef __attribute__((ext_vector_type(8))) float v8f;

#if __has_builtin(__builtin_amdgcn_global_load_async_to_lds_b128) && \
    __has_builtin(__builtin_amdgcn_s_wait_asynccnt)
#define USE_ASYNC_LDS 1
typedef __attribute__((__vector_size__(16))) int v4i_t;
typedef __attribute__((address_space(1))) v4i_t as1_v4i;  // prints as __device__
typedef __attribute__((address_space(3))) v4i_t as3_v4i;  // prints as __shared__
__device__ __forceinline__ void async_copy_b128(const void* g, void* l) {
  // flat->AS1 is an identity on the address bits; a flat shared pointer keeps
  // the LDS offset in its low 32 bits, so the integer round-trip yields AS3.
  __builtin_amdgcn_global_load_async_to_lds_b128(
      (as1_v4i*)(unsigned long long)g,
      (as3_v4i*)(unsigned)(unsigned long long)l, 0, 0);
}
#else
#define USE_ASYNC_LDS 0
#endif

// ---- kernel 1: row norms + dual coefficients --------------------------------
__global__ void svm_precompute(const float* __restrict__ X,
                               const float* __restrict__ Xt,
                               const float* __restrict__ alphas,
                               const float* __restrict__ y,
                               float* __restrict__ xsq,
                               float* __restrict__ xtsq,
                               float* __restrict__ coef,
                               int N, int M) {
  const int lane = threadIdx.x & 31;
  const int row  = blockIdx.x * 8 + (threadIdx.x >> 5);  // one wave per row
  if (row >= N + M) return;
  const float* src = (row < N) ? (X + (size_t)row * D_DIM)
                               : (Xt + (size_t)(row - N) * D_DIM);
  float s = 0.f;
#pragma unroll
  for (int i = 0; i < D_DIM / 32; ++i) {
    float v = src[lane + 32 * i];
    s += v * v;
  }
#pragma unroll
  for (int m = 16; m >= 1; m >>= 1) s += __shfl_xor(s, m, 32);
  if (lane == 0) {
    if (row < N) {
      xsq[row] = s;
    } else {
      const int r = row - N;
      xtsq[r] = s;
      coef[r] = alphas[r] * y[r];
    }
  }
}

// ---- tile staging: 4 X_train tiles (64 KB) into one LDS buffer --------------
__device__ __forceinline__ void stage_tiles(const float* __restrict__ Xt,
                                            float* __restrict__ dst,
                                            int t, int tid) {
#pragma unroll
  for (int i = 0; i < (CGRPS * TILE_N * D_DIM / 4) / 256; ++i) {  // 16/thread
    int idx = tid + 256 * i;
    int cg  = idx >> 10;          // 1024 float4 per tile
    int rem = idx & 1023;
    int n   = rem >> 6;
    int kc  = (rem & 63) << 2;
    int j   = (cg + CGRPS * t) * TILE_N + n;
    const float* g = Xt + (size_t)j * D_DIM + kc;
    float*       l = dst + (cg * TILE_N + n) * LDS_STRIDE + kc;
#if USE_ASYNC_LDS
    async_copy_b128(g, l);        // no VGPR data, tracked by ASYNCcnt
#else
    *(float4*)l = *(const float4*)g;
#endif
  }
}

__device__ __forceinline__ void wait_stage_done() {
#if USE_ASYNC_LDS
  __builtin_amdgcn_s_wait_asynccnt(0);
#endif
  __syncthreads();
}

// ---- kernel 2: fused WMMA GEMM + RBF epilogue -------------------------------
__global__ void __launch_bounds__(256)
svm_rbf_wmma(const float* __restrict__ X,
             const float* __restrict__ Xt,
             const float* __restrict__ xsq,
             const float* __restrict__ xtsq,
             const float* __restrict__ coef,
             const float* __restrict__ bptr,
             float* __restrict__ out,
             int M) {
  extern __shared__ float smem[];
  float* Xs     = smem;                              // 32 x 260 f32
  float* XtS0   = Xs + ROWS_PER_BLOCK * LDS_STRIDE;  // buffer 0: 4 x 16 x 260
  float* XtS1   = XtS0 + TILE_FLOATS;                // buffer 1
  float* outAcc = XtS1 + TILE_FLOATS;                // 32 f32

  const int tid  = threadIdx.x;
  const int lane = tid & 31;
  const int wave = tid >> 5;
  const int rgrp = wave & 1;   // which 16-row half of the block
  const int cgrp = wave >> 1;  // which of 4 column groups
  const int g    = lane >> 4;  // lane half (selects K pair / M+8)
  const int m16  = lane & 15;  // M index (A) / N index (B, C)
  const int r0   = blockIdx.x * ROWS_PER_BLOCK;

  // stage the block's 32 X rows into LDS (8 float4 per thread)
#pragma unroll
  for (int i = 0; i < (ROWS_PER_BLOCK * D_DIM / 4) / 256; ++i) {
    int idx = tid + 256 * i;
    int row = idx >> 6;            // 64 float4 per row
    int kc  = (idx & 63) << 2;
    float4 v = *(const float4*)(X + (size_t)(r0 + row) * D_DIM + kc);
    *(float4*)(Xs + row * LDS_STRIDE + kc) = v;
  }
  if (tid < ROWS_PER_BLOCK) outAcc[tid] = 0.f;

  // per-lane ||x||^2 for the 8 C-matrix rows this lane holds
  float xsqv[8];
#pragma unroll
  for (int vg = 0; vg < 8; ++vg)
    xsqv[vg] = xsq[r0 + rgrp * 16 + vg + 8 * g];

  const float* Abase = Xs + rgrp * TILE_N * LDS_STRIDE;
  const int fragOff  = m16 * LDS_STRIDE + 2 * g;  // lane g*16+m: K = 2g + vgpr

  float psum[8];
#pragma unroll
  for (int vg = 0; vg < 8; ++vg) psum[vg] = 0.f;

  const int numTiles = M / (TILE_N * CGRPS);  // 128 iterations x 4 tiles

  stage_tiles(Xt, XtS0, 0, tid);  // prologue: fill buffer 0

  for (int t = 0; t < numTiles; ++t) {
    wait_stage_done();  // buffer (t&1) filled; previous compute finished

    // kick off async fill of the other buffer for tile t+1
    if (t + 1 < numTiles)
      stage_tiles(Xt, ((t + 1) & 1) ? XtS1 : XtS0, t + 1, tid);

    // prefetch epilogue scalars early (used ~64 WMMAs later)
    const int j0 = (cgrp + CGRPS * t) * TILE_N;
    const float xtsq_n = xtsq[j0 + m16];
    const float coef_n = coef[j0 + m16];

    // 16x16 fp32 tile: S = A * B^T over K = 256, 64 chained WMMAs
    const float* Bbase = ((t & 1) ? XtS1 : XtS0) + cgrp * TILE_N * LDS_STRIDE;
    v8f acc = {0.f, 0.f, 0.f, 0.f, 0.f, 0.f, 0.f, 0.f};
#pragma unroll 8
    for (int k0 = 0; k0 < D_DIM; k0 += 4) {
      v2f a = *(const v2f*)(Abase + fragOff + k0);  // one ds_load_b64 each
      v2f b = *(const v2f*)(Bbase + fragOff + k0);
      acc = __builtin_amdgcn_wmma_f32_16x16x4_f32(false, a, false, b,
                                                  (short)0, acc, false, false);
    }

    // fused RBF epilogue: exp on the TRANS pipe, accumulate per lane
#pragma unroll
    for (int vg = 0; vg < 8; ++vg) {
      float d = xsqv[vg] + xtsq_n - 2.f * acc[vg];
      psum[vg] += __expf(-GAMMA * d) * coef_n;
    }
  }

  // reduce over the 16 lanes of each half-wave, then across waves via LDS
#pragma unroll
  for (int vg = 0; vg < 8; ++vg) {
    float v = psum[vg];
    v += __shfl_xor(v, 1, 32);
    v += __shfl_xor(v, 2, 32);
    v += __shfl_xor(v, 4, 32);
    v += __shfl_xor(v, 8, 32);
    if (m16 == 0)  // lanes 0 (rows vg) and 16 (rows vg+8) survive
      atomicAdd(&outAcc[rgrp * 16 + vg + 8 * g], v);
  }
  __syncthreads();
  if (tid < ROWS_PER_BLOCK) out[r0 + tid] = outAcc[tid] + bptr[0];
}

// ---------------------------------------------------------------------------
extern "C" void kernel_launch(void* const* d_in, const int* in_sizes, int n_in,
                              void* d_out, int out_size, void* d_ws, size_t ws_size,
                              hipStream_t stream) {
  const float* X      = (const float*)d_in[0];
  const float* Xt     = (const float*)d_in[1];
  const float* alphas = (const float*)d_in[2];
  const float* y      = (const float*)d_in[3];
  const float* b      = (const float*)d_in[4];
  const int N = in_sizes[0] / D_DIM;
  const int M = in_sizes[2];

  float* ws   = (float*)d_ws;
  float* xsq  = ws;
  float* xtsq = ws + N;
  float* coef = ws + N + M;

  svm_precompute<<<(N + M + 7) / 8, 256, 0, stream>>>(X, Xt, alphas, y,
                                                      xsq, xtsq, coef, N, M);

  const size_t ldsBytes =
      (size_t)(ROWS_PER_BLOCK * LDS_STRIDE + 2 * TILE_FLOATS + ROWS_PER_BLOCK) *
      sizeof(float);  // ~163 KB of the 320 KB WGP LDS
  svm_rbf_wmma<<<N / ROWS_PER_BLOCK, 256, ldsBytes, stream>>>(
      X, Xt, xsq, xtsq, coef, b, (float*)d_out, M);
}